// MetricLoss_12687333392690
// MI455X (gfx1250) — compile-verified
//
#include <hip/hip_runtime.h>
#include <hip/hip_bf16.h>

#define NBLK 1024
#define TPB  256

#define LOG2E_F   1.4426950408889634f
#define LN2_F     0.6931471805599453f
#define NHALF_LOG2E_F (-0.7213475204444817f)   // -0.5 * log2(e)

// native gfx1250 trans ops (single v_exp/v_log/v_rcp/v_sqrt instructions)
__device__ __forceinline__ float fexp2(float x)  { return __builtin_amdgcn_exp2f(x); }
__device__ __forceinline__ float flog2(float x)  { return __builtin_amdgcn_logf(x); }
__device__ __forceinline__ float frcp(float x)   { return __builtin_amdgcn_rcpf(x); }
__device__ __forceinline__ float fsqrt(float x)  { return __builtin_amdgcn_sqrtf(x); }

// Per-row loss for symmetric 2x2 packed as (a,b,c) = [[a,b],[b,c]].
// Key fusion: T = expm(P)^(-1/2) = expm(-P/2)  (matrix functions compose),
// so T comes straight from P's eigendecomposition -> one eig2+recon saved.
__device__ __forceinline__ float row_loss(float pa, float pb, float pc,
                                          float ma, float mb, float mc) {
    // ---- eig of P ----
    float mid  = 0.5f * (pa + pc);
    float hd   = 0.5f * (pa - pc);
    float disc = fsqrt(fmaf(hd, hd, pb * pb));
    float l1   = mid - disc;
    float l2   = mid + disc;

    // ---- T = expm(-P/2):  f(l) = exp(-l/2) = exp2(l * -0.5*log2e) ----
    float f1   = fexp2(l1 * NHALF_LOG2E_F);
    float f2   = fexp2(l2 * NHALF_LOG2E_F);
    float coef = (f2 - f1) * frcp(fmaxf(2.0f * disc, 1e-12f));  // l2-l1 == 2*disc
    float ta   = fmaf(coef, pa - l1, f1);
    float tb   = coef * pb;
    float tc   = fmaf(coef, pc - l1, f1);

    // ---- Pm = T * M * T (symmetric congruence) ----
    float u00 = fmaf(ma, ta, mb * tb);
    float u01 = fmaf(ma, tb, mb * tc);
    float u10 = fmaf(mb, ta, mc * tb);
    float u11 = fmaf(mb, tb, mc * tc);
    float p00 = fmaf(ta, u00, tb * u10);
    float p01 = fmaf(ta, u01, tb * u11);
    float p11 = fmaf(tb, u01, tc * u11);

    // ---- S = logm(Pm) ----
    mid  = 0.5f * (p00 + p11);
    hd   = 0.5f * (p00 - p11);
    disc = fsqrt(fmaf(hd, hd, p01 * p01));
    l1   = mid - disc;
    l2   = mid + disc;
    f1   = flog2(l1) * LN2_F;
    f2   = flog2(l2) * LN2_F;
    coef = (f2 - f1) * frcp(fmaxf(2.0f * disc, 1e-12f));
    float sa = fmaf(coef, p00 - l1, f1);
    float sb = coef * p01;
    float sc = fmaf(coef, p11 - l1, f1);

    // sa^2 + 2*sb^2 + sc^2
    return fmaf(sa, sa, fmaf(2.0f * sb, sb, sc * sc));
}

// ---- pass 1: per-block partial sums ---------------------------------------

__global__ __launch_bounds__(TPB) void metric_partial_kernel(
        const float* __restrict__ pred, const float* __restrict__ act,
        float* __restrict__ partials, int nquad, int n) {
    const float4* __restrict__ p4 = (const float4*)pred;
    const float4* __restrict__ a4 = (const float4*)act;

    const int tid    = blockIdx.x * blockDim.x + threadIdx.x;
    const int stride = gridDim.x * blockDim.x;

    float acc = 0.0f;

    // 4 rows (12 floats = 3 x float4) per thread per iteration:
    // fully coalesced global_load_b128 stream.
    for (int q = tid; q < nquad; q += stride) {
        float4 pA = p4[3 * q + 0];
        float4 pB = p4[3 * q + 1];
        float4 pC = p4[3 * q + 2];
        float4 aA = a4[3 * q + 0];
        float4 aB = a4[3 * q + 1];
        float4 aC = a4[3 * q + 2];

        // speculative prefetch of next grid-stride chunk (global_prefetch_b8)
        __builtin_prefetch(&p4[3 * (q + stride)], 0, 0);
        __builtin_prefetch(&a4[3 * (q + stride)], 0, 0);

        acc += row_loss(pA.x, pA.y, pA.z, aA.x, aA.y, aA.z);
        acc += row_loss(pA.w, pB.x, pB.y, aA.w, aB.x, aB.y);
        acc += row_loss(pB.z, pB.w, pC.x, aB.z, aB.w, aC.x);
        acc += row_loss(pC.y, pC.z, pC.w, aC.y, aC.z, aC.w);
    }

    // tail rows if n % 4 != 0 (not hit for N = 4194304, kept for safety)
    for (int r = nquad * 4 + tid; r < n; r += stride) {
        acc += row_loss(pred[3 * r], pred[3 * r + 1], pred[3 * r + 2],
                        act[3 * r],  act[3 * r + 1],  act[3 * r + 2]);
    }

    __shared__ float sm[TPB];
    sm[threadIdx.x] = acc;
    __syncthreads();
#pragma unroll
    for (int off = TPB / 2; off > 0; off >>= 1) {
        if (threadIdx.x < off) sm[threadIdx.x] += sm[threadIdx.x + off];
        __syncthreads();
    }
    if (threadIdx.x == 0) partials[blockIdx.x] = sm[0];
}

// ---- pass 2: fold partials, write mean ------------------------------------

__global__ __launch_bounds__(1024) void metric_final_kernel(
        const float* __restrict__ partials, float* __restrict__ out,
        int nparts, float inv_n) {
    __shared__ float sm[1024];
    float v = (threadIdx.x < (unsigned)nparts) ? partials[threadIdx.x] : 0.0f;
    sm[threadIdx.x] = v;
    __syncthreads();
#pragma unroll
    for (int off = 512; off > 0; off >>= 1) {
        if (threadIdx.x < off) sm[threadIdx.x] += sm[threadIdx.x + off];
        __syncthreads();
    }
    if (threadIdx.x == 0) out[0] = sm[0] * inv_n;
}

extern "C" void kernel_launch(void* const* d_in, const int* in_sizes, int n_in,
                              void* d_out, int out_size, void* d_ws, size_t ws_size,
                              hipStream_t stream) {
    const float* pred = (const float*)d_in[0];   // (N,3) float32
    const float* act  = (const float*)d_in[1];   // (N,3) float32
    float* out        = (float*)d_out;           // scalar float32
    float* partials   = (float*)d_ws;            // NBLK floats of scratch

    const int n     = in_sizes[0] / 3;           // rows
    const int nquad = n / 4;                     // row-quads (3 float4 each)
    const float inv_n = 1.0f / (float)n;

    metric_partial_kernel<<<NBLK, TPB, 0, stream>>>(pred, act, partials, nquad, n);
    metric_final_kernel<<<1, 1024, 0, stream>>>(partials, out, NBLK, inv_n);
}